// TaskAlignedAssigner_9483287789832
// MI455X (gfx1250) — compile-verified
//
#include <hip/hip_runtime.h>
#include <hip/hip_bf16.h>

#define TOPK 13
#define PI_F 3.14159265358979323846f

typedef __attribute__((ext_vector_type(2))) float v2f;
typedef __attribute__((ext_vector_type(8))) float v8f;

// Exact circle-IoU per the reference formulas (f32).
__device__ __forceinline__ float circle_iou(float d, float r0, float r1) {
    float rsum = r0 + r1;
    float rdif = fabsf(r0 - r1);
    float inter;
    if (d >= rsum) {
        inter = 0.0f;
    } else if (d <= rdif) {
        float mr = fminf(r0, r1);
        inter = PI_F * mr * mr;
    } else {
        float d2 = d * d, r02 = r0 * r0, r12 = r1 * r1;
        float t0 = fminf(1.0f, fmaxf(-1.0f, (d2 + r02 - r12) / (2.0f * d * r0)));
        float t1 = fminf(1.0f, fmaxf(-1.0f, (d2 + r12 - r02) / (2.0f * d * r1)));
        float pr = (-d + r0 + r1) * (d + r0 - r1) * (d - r0 + r1) * (d + r0 + r1);
        float sq = sqrtf(fmaxf(0.0f, pr));
        inter = r02 * acosf(t0) + r12 * acosf(t1) - 0.5f * sq;
    }
    float uni = PI_F * r0 * r0 + PI_F * r1 * r1 - inter;
    return (uni > 0.0f) ? inter / uni : 0.0f;
}

// top_k tie-break: higher value wins; equal value -> lower index wins (lax.top_k stability)
__device__ __forceinline__ bool better(float v, int i, float V, int I) {
    return (v > V) || (v == V && i < I);
}

// ---------------- Kernel Z: zero the selection bitmask ----------------
__global__ void zero_kernel(unsigned long long* __restrict__ p, size_t n) {
    size_t i = (size_t)blockIdx.x * blockDim.x + threadIdx.x;
    if (i < n) p[i] = 0ull;
}

// ---------------- Kernel M: anchor-in-gt bitmask via f32 WMMA ----------------
// d^2[g][a] = (-2gx)*ax + (-2gy)*ay + 1*(ax^2+ay^2) + (gx^2+gy^2)*1
// A (16x4 f32): lane L -> row M=L%16; lanes<16 hold K=0,1; lanes>=16 hold K=2,3
// B (4x16 f32): lane L -> col N=L%16; lanes<16 hold K=0,1; lanes>=16 hold K=2,3
// D (16x16 f32): lane L, vgpr v -> D[v + 8*(L>=16)][L%16]
__global__ void ingt_kernel(const float* __restrict__ anc,
                            const float* __restrict__ gt,
                            unsigned long long* __restrict__ inGt,
                            int na, int nb) {
    const int b    = blockIdx.y;
    const int tid  = threadIdx.x;
    const int lane = tid & 31;
    const int wave = tid >> 5;
    const int half = lane >> 4;
    const int col  = lane & 15;

    __shared__ float sAx[64];  // -2*gx
    __shared__ float sAy[64];  // -2*gy
    __shared__ float sN[64];   // gx^2+gy^2
    __shared__ float sRR[64];  // r^2 (or -1 for padded slots)
    for (int g = tid; g < 64; g += blockDim.x) {
        if (g < nb) {
            float gx = gt[((size_t)b * nb + g) * 3 + 0];
            float gy = gt[((size_t)b * nb + g) * 3 + 1];
            float r  = gt[((size_t)b * nb + g) * 3 + 2];
            sAx[g] = -2.0f * gx;
            sAy[g] = -2.0f * gy;
            sN[g]  = gx * gx + gy * gy;
            sRR[g] = r * r;
        } else {
            sAx[g] = 0.0f; sAy[g] = 0.0f; sN[g] = 0.0f; sRR[g] = -1.0f;
        }
    }
    __syncthreads();

    const int wavesPerBlock = blockDim.x >> 5;
    const int tile  = blockIdx.x * wavesPerBlock + wave;
    const int aBase = tile * 16;
    if (aBase >= na) return;  // wave-uniform exit: EXEC stays all-1 for WMMA

    const int a = aBase + col;
    const bool aValid = (a < na);
    float ax = aValid ? anc[2 * a + 0] : 0.0f;
    float ay = aValid ? anc[2 * a + 1] : 0.0f;
    float an = aValid ? (ax * ax + ay * ay) : 3.0e38f;  // huge d^2 -> never inside

    v2f Bop;
    Bop.x = half ? an   : ax;
    Bop.y = half ? 1.0f : ay;

    // Compact bitmask: bit (t*8 + v) set <=> gt (t*16 + 8*half + v) contains this anchor.
    // Built with 32-bit ops (static shift per v), expanded to 64-bit once at the end.
    unsigned compact = 0u;
    const int nTiles = (nb + 15) >> 4;  // <= 4
    for (int t = 0; t < nTiles; ++t) {
        int gA = t * 16 + col;
        v2f Aop;
        Aop.x = half ? 1.0f    : sAx[gA];
        Aop.y = half ? sN[gA]  : sAy[gA];
        v8f c = {};
        c = __builtin_amdgcn_wmma_f32_16x16x4_f32(
            /*neg_a=*/false, Aop, /*neg_b=*/false, Bop,
            /*c_mod=*/(short)0, c, /*reuse_a=*/false, /*reuse_b=*/false);
        const unsigned tb = 1u << (t * 8);
        const float* rr = &sRR[t * 16 + 8 * half];
#pragma unroll
        for (int v = 0; v < 8; ++v) {
            if (c[v] < rr[v]) compact |= (tb << v);  // rad - d > EPS  <=>  d^2 < r^2
        }
    }

    // Expand compact bytes (one per tile) to bit positions t*16, then offset by half.
    unsigned long long bits =
          (unsigned long long)(compact & 0x000000FFu)
        | ((unsigned long long)(compact & 0x0000FF00u) << 8)
        | ((unsigned long long)(compact & 0x00FF0000u) << 16)
        | ((unsigned long long)(compact & 0xFF000000u) << 24);
    bits <<= (8 * half);

    // Merge the two half-wave bit sets (disjoint bit positions) for anchor `col`.
    unsigned lo  = (unsigned)(bits & 0xffffffffull);
    unsigned hi  = (unsigned)(bits >> 32);
    unsigned plo = (unsigned)__shfl_xor((int)lo, 16, 32);
    unsigned phi = (unsigned)__shfl_xor((int)hi, 16, 32);
    bits = (((unsigned long long)(hi | phi)) << 32) | (unsigned long long)(lo | plo);
    if (half == 0 && aValid) inGt[(size_t)b * na + a] = bits;
}

// ---------------- Kernel A: per-(b,gt) top-13 anchors by align metric ----------------
__global__ void __launch_bounds__(256)
topk_kernel(const float* __restrict__ pd,
            const unsigned long long* __restrict__ inGt,
            const float* __restrict__ gt,
            const float* __restrict__ mgt,
            int* __restrict__ topkIdx,
            int na, int nb) {
    const int g = blockIdx.x;
    const int b = blockIdx.y;
    if (mgt[(size_t)b * nb + g] <= 0.0f) return;  // masked gts discarded downstream
    const int tid = threadIdx.x;

    const float gx = gt[((size_t)b * nb + g) * 3 + 0];
    const float gy = gt[((size_t)b * nb + g) * 3 + 1];
    const float gr = gt[((size_t)b * nb + g) * 3 + 2];

    float lv[TOPK];
    int   li[TOPK];
#pragma unroll
    for (int k = 0; k < TOPK; ++k) { lv[k] = -1.0f; li[k] = 0x7fffffff; }

    const unsigned long long* wrow = inGt + (size_t)b * na;
    const float* pdB = pd + (size_t)b * na * 3;

    for (int a = tid; a < na; a += blockDim.x) {
        int pf = a + 8 * 256; if (pf >= na) pf = na - 1;
        __builtin_prefetch(&wrow[pf], 0, 1);
        unsigned long long w = wrow[a];
        float metric = 0.0f;
        if ((w >> g) & 1ull) {
            float px = pdB[(size_t)a * 3 + 0];
            float py = pdB[(size_t)a * 3 + 1];
            float pr = pdB[(size_t)a * 3 + 2];
            float dx = gx - px, dy = gy - py;
            float d = sqrtf(dx * dx + dy * dy);
            float rsum = gr + pr;
            float dist = fmaxf(0.0f, (rsum - d) / rsum);
            float iou = circle_iou(d, gr, pr);
            float i2 = iou * iou;
            float i6 = (i2 * iou) * (i2 * iou);
            metric = dist * i6;  // ALPHA=1, BETA=6
        }
        if (better(metric, a, lv[TOPK - 1], li[TOPK - 1])) {
            float cv = metric; int ci = a;
#pragma unroll
            for (int k = 0; k < TOPK; ++k) {
                if (better(cv, ci, lv[k], li[k])) {
                    float tv = lv[k]; int ti = li[k];
                    lv[k] = cv; li[k] = ci; cv = tv; ci = ti;
                }
            }
        }
    }

    __shared__ float sv[256 * TOPK];
    __shared__ int   si[256 * TOPK];
    __shared__ float rv[256];
    __shared__ int   ri[256];
    __shared__ int   ro[256];
#pragma unroll
    for (int k = 0; k < TOPK; ++k) { sv[tid * TOPK + k] = lv[k]; si[tid * TOPK + k] = li[k]; }
    __syncthreads();

    for (int r = 0; r < TOPK; ++r) {
        float bv = -2.0f; int bi = 0x7fffffff; int bo = -1;
#pragma unroll
        for (int k = 0; k < TOPK; ++k) {
            float v = sv[tid * TOPK + k]; int i = si[tid * TOPK + k];
            if (better(v, i, bv, bi)) { bv = v; bi = i; bo = tid * TOPK + k; }
        }
        rv[tid] = bv; ri[tid] = bi; ro[tid] = bo;
        __syncthreads();
        for (int s = blockDim.x >> 1; s > 0; s >>= 1) {
            if (tid < s) {
                if (better(rv[tid + s], ri[tid + s], rv[tid], ri[tid])) {
                    rv[tid] = rv[tid + s]; ri[tid] = ri[tid + s]; ro[tid] = ro[tid + s];
                }
            }
            __syncthreads();
        }
        if (tid == 0) {
            topkIdx[((size_t)b * nb + g) * TOPK + r] = ri[0];
            if (ro[0] >= 0) { sv[ro[0]] = -3.0f; si[ro[0]] = 0x7fffffff; }
        }
        __syncthreads();
    }
}

// ---------------- Kernel B: scatter top-13 indices into 64-bit selection mask ----------------
__global__ void scatter_kernel(const int* __restrict__ topkIdx,
                               const float* __restrict__ mgt,
                               unsigned long long* __restrict__ selMask,
                               int na, int nb, int total) {
    int i = blockIdx.x * blockDim.x + threadIdx.x;
    if (i >= total) return;
    int bg = i / TOPK;
    int g  = bg % nb;
    int b  = bg / nb;
    if (mgt[bg] <= 0.0f) return;  // masked gt -> contributes nothing (count>1 trick)
    int a = topkIdx[i];
    if ((unsigned)a < (unsigned)na)
        atomicOr(&selMask[(size_t)b * na + a], 1ull << g);
}

// ---------------- Kernel C: per-anchor resolution + outputs ----------------
__global__ void __launch_bounds__(256)
finalize_kernel(const float* __restrict__ pd,
                const float* __restrict__ gt,
                const float* __restrict__ mgt,
                const unsigned long long* __restrict__ inGt,
                const unsigned long long* __restrict__ selMask,
                float* __restrict__ out,
                int bs, int na, int nb) {
    const int b = blockIdx.y;
    const int a = blockIdx.x * blockDim.x + threadIdx.x;

    __shared__ float sgx[64], sgy[64], sgr[64];
    __shared__ unsigned svlo, svhi;
    if (threadIdx.x == 0) { svlo = 0u; svhi = 0u; }
    __syncthreads();
    for (int g = threadIdx.x; g < nb; g += blockDim.x) {
        sgx[g] = gt[((size_t)b * nb + g) * 3 + 0];
        sgy[g] = gt[((size_t)b * nb + g) * 3 + 1];
        sgr[g] = gt[((size_t)b * nb + g) * 3 + 2];
        if (mgt[(size_t)b * nb + g] > 0.0f)
            atomicOr((g < 32) ? &svlo : &svhi, 1u << (g & 31));
    }
    __syncthreads();
    if (a >= na) return;
    unsigned long long vm = (((unsigned long long)svhi) << 32) | (unsigned long long)svlo;

    unsigned long long ing = inGt[(size_t)b * na + a];
    unsigned long long pos = selMask[(size_t)b * na + a] & ing;  // sel already limited to valid gts
    int fg = __popcll(pos);

    int tgt; float fgm;
    if (fg == 0) {
        tgt = 0; fgm = 0.0f;
    } else if (fg == 1) {
        tgt = __ffsll((long long)pos) - 1; fgm = 1.0f;
    } else {
        // argmax over g of overlaps (= iou where mask_valid, else 0); first-max wins;
        // if max overlap is 0, argmax over the full (all-zero-elsewhere) row is index 0.
        float px = pd[((size_t)b * na + a) * 3 + 0];
        float py = pd[((size_t)b * na + a) * 3 + 1];
        float pr = pd[((size_t)b * na + a) * 3 + 2];
        unsigned long long m = ing & vm;
        float bv = -1.0f; int bg2 = 0;
        while (m) {
            int g2 = __ffsll((long long)m) - 1;
            m &= m - 1;
            float dx = sgx[g2] - px, dy = sgy[g2] - py;
            float d = sqrtf(dx * dx + dy * dy);
            float ov = circle_iou(d, sgr[g2], pr);
            if (ov > bv) { bv = ov; bg2 = g2; }
        }
        if (bv <= 0.0f) bg2 = 0;
        tgt = bg2; fgm = 1.0f;
    }

    size_t o0 = ((size_t)b * na + a) * 3;
    out[o0 + 0] = sgx[tgt];
    out[o0 + 1] = sgy[tgt];
    out[o0 + 2] = sgr[tgt];
    size_t base1 = (size_t)bs * na * 3;
    out[base1 + (size_t)b * na + a] = fgm;                     // fg_mask (bool as float)
    out[base1 + (size_t)bs * na + (size_t)b * na + a] = (float)tgt;  // target_gt_idx
}

extern "C" void kernel_launch(void* const* d_in, const int* in_sizes, int n_in,
                              void* d_out, int out_size, void* d_ws, size_t ws_size,
                              hipStream_t stream) {
    const float* pd  = (const float*)d_in[0];   // (bs, na, 3)
    const float* anc = (const float*)d_in[1];   // (na, 2)
    // d_in[2] = gt_labels (int64) — unused by the reference outputs
    const float* gt  = (const float*)d_in[3];   // (bs, nb, 3)
    const float* mgt = (const float*)d_in[4];   // (bs, nb, 1)

    const int na = in_sizes[1] / 2;
    const int bs = in_sizes[0] / (3 * na);
    const int nb = in_sizes[4] / bs;            // assumed <= 64 (bitmask)

    unsigned long long* inGt    = (unsigned long long*)d_ws;
    unsigned long long* selMask = inGt + (size_t)bs * na;
    int* topkIdx = (int*)(selMask + (size_t)bs * na);
    float* out = (float*)d_out;

    // Z: zero selection mask
    {
        size_t n = (size_t)bs * na;
        int blocks = (int)((n + 255) / 256);
        zero_kernel<<<blocks, 256, 0, stream>>>(selMask, n);
    }
    // M: WMMA anchor-in-gt bitmask
    {
        int tiles = (na + 15) / 16;
        int blocks = (tiles + 7) / 8;  // 8 waves / block, 16 anchors / wave
        dim3 grid(blocks, bs);
        ingt_kernel<<<grid, 256, 0, stream>>>(anc, gt, inGt, na, nb);
    }
    // A: per-gt top-13
    {
        dim3 grid(nb, bs);
        topk_kernel<<<grid, 256, 0, stream>>>(pd, inGt, gt, mgt, topkIdx, na, nb);
    }
    // B: scatter selections
    {
        int total = bs * nb * TOPK;
        scatter_kernel<<<(total + 255) / 256, 256, 0, stream>>>(topkIdx, mgt, selMask, na, nb, total);
    }
    // C: finalize outputs
    {
        dim3 grid((na + 255) / 256, bs);
        finalize_kernel<<<grid, 256, 0, stream>>>(pd, gt, mgt, inGt, selMask, out, bs, na, nb);
    }
}